// QPNF_48103633715416
// MI455X (gfx1250) — compile-verified
//
#include <hip/hip_runtime.h>
#include <math.h>

// ---------------- problem constants ----------------
#define N_   8192
#define Dd   32
#define Hh   128
#define Mm   8
#define NYC  (Dd + 1)           // 33 columns of state
#define NY   (N_ * NYC)         // 270336
#define LW   132                // padded LDS stride for 128-wide rows
#define LD   36                 // padded LDS stride for 32-wide rows

// precomputed-constants region layout (floats)
#define PRE_E     0             // E[j][i] = W2[j][i] * (W1 IP W3)[i][j]   (128x128)
#define PRE_M3T   16384         // M3t = IP @ W3  (32x128)
#define PRE_P     20480         // P (32x32)
#define PRE_C3    21504         // c3 = b3 @ IP (32)
#define PRE_TRP   21536         // trace(P)
#define PRE_FLOATS 21568

typedef __attribute__((ext_vector_type(2))) float v2f;
typedef __attribute__((ext_vector_type(8))) float v8f;

// --------- multi-tile WMMA GEMM: acc[ct](16x16) += A(16xK) * Bt(ct-tile)(16xK)^T
// k0 is the OUTER loop so the A fragment is loaded once and reused for all NT
// column tiles; consecutive v_wmma target different accumulators (no RAW chain).
template <bool MASK_A, int K, int NT>
__device__ __forceinline__ void wmma_gemm_multi(const float* __restrict__ A, int lda,
                                                const float* __restrict__ Bt, int ldb,
                                                v8f* acc, int row, int kh) {
#pragma unroll
  for (int k0 = 0; k0 < K; k0 += 4) {
    float a0 = A[row * lda + k0 + kh];
    float a1 = A[row * lda + k0 + kh + 1];
    if (MASK_A) { a0 = (a0 > 0.f) ? 1.f : 0.f; a1 = (a1 > 0.f) ? 1.f : 0.f; }
    v2f av; av[0] = a0; av[1] = a1;
#pragma unroll
    for (int ct = 0; ct < NT; ++ct) {
      v2f bv;
      bv[0] = Bt[(ct * 16 + row) * ldb + k0 + kh];
      bv[1] = Bt[(ct * 16 + row) * ldb + k0 + kh + 1];
      acc[ct] = __builtin_amdgcn_wmma_f32_16x16x4_f32(false, av, false, bv,
                                                      (short)0, acc[ct], false, false);
    }
  }
}

// coalesced float4 global -> padded LDS staging (cols divisible by 4)
__device__ __forceinline__ void stage_padded(const float* __restrict__ g, float* s,
                                             int rows, int cols, int ldp, int tid) {
  int n4 = (rows * cols) >> 2;
  const float4* src = (const float4*)g;
  for (int i = tid; i < n4; i += 128) {
    int e = i << 2;
    int r = e / cols;
    int c = e - r * cols;
    *(float4*)(&s[r * ldp + c]) = src[i];
  }
}

// ---------------- precompute kernel ----------------
__global__ __launch_bounds__(256) void precompute_kernel(
    const float* __restrict__ Gw, const float* __restrict__ W1,
    const float* __restrict__ W2, const float* __restrict__ W3,
    const float* __restrict__ b3, float* __restrict__ pre) {
  __shared__ float sGw[Mm * Dd];
  __shared__ float sA[Mm * Mm];
  __shared__ float sAinvG[Mm * Dd];
  __shared__ float sIP[Dd * Dd];
  __shared__ float sT[Hh * Dd];       // W1 @ IP
  __shared__ float sW3[Dd * Hh];
  int tid = threadIdx.x;
  for (int i = tid; i < Mm * Dd; i += 256) sGw[i] = Gw[i];
  for (int i = tid; i < Dd * Hh; i += 256) sW3[i] = W3[i];
  __syncthreads();
  if (tid < Mm * Mm) {
    int i = tid / Mm, j = tid % Mm;
    float s = 0.f;
    for (int k = 0; k < Dd; ++k) s += sGw[i * Dd + k] * sGw[j * Dd + k];
    sA[tid] = s;
  }
  __syncthreads();
  if (tid == 0) {  // pivoted Gauss-Jordan: A X = Gw  ->  X = AinvG
    for (int i = 0; i < Mm * Dd; ++i) sAinvG[i] = sGw[i];
    float Al[Mm][Mm];
    for (int i = 0; i < Mm; ++i) for (int j = 0; j < Mm; ++j) Al[i][j] = sA[i * Mm + j];
    for (int c = 0; c < Mm; ++c) {
      int p = c; float best = fabsf(Al[c][c]);
      for (int r = c + 1; r < Mm; ++r) { float v = fabsf(Al[r][c]); if (v > best) { best = v; p = r; } }
      if (p != c) {
        for (int j = 0; j < Mm; ++j) { float t = Al[c][j]; Al[c][j] = Al[p][j]; Al[p][j] = t; }
        for (int j = 0; j < Dd; ++j) { float t = sAinvG[c*Dd+j]; sAinvG[c*Dd+j] = sAinvG[p*Dd+j]; sAinvG[p*Dd+j] = t; }
      }
      float inv = 1.f / Al[c][c];
      for (int j = c; j < Mm; ++j) Al[c][j] *= inv;
      for (int j = 0; j < Dd; ++j) sAinvG[c * Dd + j] *= inv;
      for (int r = 0; r < Mm; ++r) if (r != c) {
        float f = Al[r][c];
        if (f != 0.f) {
          for (int j = c; j < Mm; ++j) Al[r][j] -= f * Al[c][j];
          for (int j = 0; j < Dd; ++j) sAinvG[r * Dd + j] -= f * sAinvG[c * Dd + j];
        }
      }
    }
  }
  __syncthreads();
  for (int i = tid; i < Dd * Dd; i += 256) {  // P, IP
    int a = i / Dd, b = i % Dd;
    float s = 0.f;
    for (int m = 0; m < Mm; ++m) s += sGw[m * Dd + a] * sAinvG[m * Dd + b];
    pre[PRE_P + i] = s;
    sIP[i] = ((a == b) ? 1.f : 0.f) - s;
  }
  __syncthreads();
  if (tid == 0) {
    float tr = 0.f;
    for (int a = 0; a < Dd; ++a) tr += 1.f - sIP[a * Dd + a];
    pre[PRE_TRP] = tr;
  }
  for (int i = tid; i < Hh * Dd; i += 256) {  // sT = W1 @ IP
    int r = i / Dd, b = i % Dd;
    float s = 0.f;
    for (int a = 0; a < Dd; ++a) s += W1[r * Dd + a] * sIP[a * Dd + b];
    sT[i] = s;
  }
  __syncthreads();
  for (int idx = tid; idx < Hh * Hh; idx += 256) {  // E = W2 .* B^T
    int j = idx / Hh, i = idx % Hh;
    float s = 0.f;
    for (int b = 0; b < Dd; ++b) s += sT[i * Dd + b] * sW3[b * Hh + j];
    pre[PRE_E + idx] = W2[idx] * s;
  }
  for (int idx = tid; idx < Dd * Hh; idx += 256) {  // M3t = IP @ W3
    int n = idx / Hh, k = idx % Hh;
    float s = 0.f;
    for (int b = 0; b < Dd; ++b) s += sIP[n * Dd + b] * sW3[b * Hh + k];
    pre[PRE_M3T + idx] = s;
  }
  for (int n = tid; n < Dd; n += 256) {             // c3 = b3 @ IP
    float s = 0.f;
    for (int b = 0; b < Dd; ++b) s += b3[b] * sIP[b * Dd + n];
    pre[PRE_C3 + n] = s;
  }
}

// ---------------- RHS kernel: kout = f(ybase + dtc*(a1 k1 + ... )) -------------
__global__ __launch_bounds__(128) void f_kernel(
    const float* __restrict__ ybase,
    const float* __restrict__ k1p, const float* __restrict__ k2p,
    const float* __restrict__ k3p, const float* __restrict__ k4p,
    const float* __restrict__ k5p,
    float a1, float a2, float a3, float a4, float a5, int nk,
    const float* __restrict__ ctrl,
    const float* __restrict__ W1, const float* __restrict__ b1,
    const float* __restrict__ W2, const float* __restrict__ b2,
    const float* __restrict__ pre, float* __restrict__ kout) {
  __shared__ __align__(16) float sW1[Hh * LD];
  __shared__ __align__(16) float sW2[Hh * LW];
  __shared__ __align__(16) float sE [Hh * LW];
  __shared__ __align__(16) float sM3[Dd * LW];
  __shared__ __align__(16) float sP [Dd * LD];
  __shared__ float sb1[Hh], sb2[Hh], sc3[Dd];
  __shared__ __align__(16) float sX [4][16 * LD];
  __shared__ __align__(16) float sH1[4][16 * LW];
  __shared__ __align__(16) float sH2[4][16 * LW];
  __shared__ float sRed[4][32][8];

  const int tid = threadIdx.x;
  stage_padded(W1, sW1, Hh, Dd, LD, tid);
  stage_padded(W2, sW2, Hh, Hh, LW, tid);
  stage_padded(pre + PRE_E, sE, Hh, Hh, LW, tid);
  stage_padded(pre + PRE_M3T, sM3, Dd, Hh, LW, tid);
  stage_padded(pre + PRE_P, sP, Dd, Dd, LD, tid);
  for (int i = tid; i < Hh; i += 128) { sb1[i] = b1[i]; sb2[i] = b2[i]; }
  for (int i = tid; i < Dd; i += 128) sc3[i] = pre[PRE_C3 + i];

  const float dtc = ctrl[2];
  const int s0 = blockIdx.x * 64;
  for (int idx = tid; idx < 64 * Dd; idx += 128) {  // stage x = y + dtc*sum(a_i k_i)
    int s = idx >> 5, c = idx & 31;
    int g = (s0 + s) * NYC + c;
    float acc = 0.f;
    if (nk > 0) acc += a1 * k1p[g];
    if (nk > 1) acc += a2 * k2p[g];
    if (nk > 2) acc += a3 * k3p[g];
    if (nk > 3) acc += a4 * k4p[g];
    if (nk > 4) acc += a5 * k5p[g];
    sX[s >> 4][(s & 15) * LD + c] = ybase[g] + dtc * acc;
  }
  __syncthreads();

  const int w = tid >> 5, lane = tid & 31;
  const int row = lane & 15;
  const int kh = (lane >> 4) << 1;
  float* X  = sX[w];
  float* H1 = sH1[w];
  float* H2 = sH2[w];

  // layer 1: h1 = relu(x W1^T + b1)   (8 column tiles, A loaded once per k-chunk)
  {
    v8f acc[8];
#pragma unroll
    for (int ct = 0; ct < 8; ++ct) {
      float bias = sb1[ct * 16 + row];
#pragma unroll
      for (int r = 0; r < 8; ++r) acc[ct][r] = bias;
    }
    wmma_gemm_multi<false, Dd, 8>(X, LD, sW1, LD, acc, row, kh);
#pragma unroll
    for (int ct = 0; ct < 8; ++ct)
#pragma unroll
      for (int r = 0; r < 8; ++r) {
        int m = (lane < 16) ? r : r + 8;
        float v = acc[ct][r];
        H1[m * LW + ct * 16 + row] = (v > 0.f) ? v : 0.f;
      }
  }
  __syncthreads();

  // layer 2: h2 = relu(h1 W2^T + b2)
  {
    v8f acc[8];
#pragma unroll
    for (int ct = 0; ct < 8; ++ct) {
      float bias = sb2[ct * 16 + row];
#pragma unroll
      for (int r = 0; r < 8; ++r) acc[ct][r] = bias;
    }
    wmma_gemm_multi<false, Hh, 8>(H1, LW, sW2, LW, acc, row, kh);
#pragma unroll
    for (int ct = 0; ct < 8; ++ct)
#pragma unroll
      for (int r = 0; r < 8; ++r) {
        int m = (lane < 16) ? r : r + 8;
        float v = acc[ct][r];
        H2[m * LW + ct * 16 + row] = (v > 0.f) ? v : 0.f;
      }
  }
  __syncthreads();

  // dlogp_n = d2_n^T E d1_n + trP   via  T = D1 @ E^T, then mask-dot with d2
  {
    v8f acc[8];
#pragma unroll
    for (int ct = 0; ct < 8; ++ct)
#pragma unroll
      for (int r = 0; r < 8; ++r) acc[ct][r] = 0.f;
    wmma_gemm_multi<true, Hh, 8>(H1, LW, sE, LW, acc, row, kh);
    float dlp[8];
#pragma unroll
    for (int r = 0; r < 8; ++r) dlp[r] = 0.f;
#pragma unroll
    for (int ct = 0; ct < 8; ++ct) {
      int j = ct * 16 + row;
#pragma unroll
      for (int r = 0; r < 8; ++r) {
        int m = (lane < 16) ? r : r + 8;
        dlp[r] += (H2[m * LW + j] > 0.f) ? acc[ct][r] : 0.f;
      }
    }
#pragma unroll
    for (int r = 0; r < 8; ++r) sRed[w][lane][r] = dlp[r];
  }
  __syncthreads();
  if (lane < 16) {
    float s = 0.f;
    if (lane < 8) { for (int l = 0; l < 16; ++l) s += sRed[w][l][lane]; }
    else          { for (int l = 0; l < 16; ++l) s += sRed[w][16 + l][lane - 8]; }
    kout[(s0 + w * 16 + lane) * NYC + Dd] = s + pre[PRE_TRP];
  }

  // dz = -(h2 @ M3t^T) - c3 - x @ P   (2 column tiles)
  {
    v8f acc[2];
#pragma unroll
    for (int ct = 0; ct < 2; ++ct) {
      float c3v = sc3[ct * 16 + row];
#pragma unroll
      for (int r = 0; r < 8; ++r) acc[ct][r] = c3v;
    }
    wmma_gemm_multi<false, Hh, 2>(H2, LW, sM3, LW, acc, row, kh);
    wmma_gemm_multi<false, Dd, 2>(X,  LD, sP,  LD, acc, row, kh);
#pragma unroll
    for (int ct = 0; ct < 2; ++ct)
#pragma unroll
      for (int r = 0; r < 8; ++r) {
        int m = (lane < 16) ? r : r + 8;
        kout[(s0 + w * 16 + m) * NYC + ct * 16 + row] = -acc[ct][r];
      }
  }
}

// ---------------- small orchestration kernels ----------------
__global__ void init_kernel(const float* __restrict__ z, float* __restrict__ y,
                            float* __restrict__ ctrl) {
  int i = blockIdx.x * blockDim.x + threadIdx.x;
  if (i < NY) {
    int n = i / NYC, c = i % NYC;
    y[i] = (c < Dd) ? z[n * Dd + c] : 0.f;
  }
  if (i == 0) { ctrl[0] = 0.f; ctrl[1] = 0.25f; }
}

__global__ void step_init_kernel(float* __restrict__ ctrl, double* __restrict__ errAcc,
                                 const float* __restrict__ T) {
  float t = ctrl[0], dt = ctrl[1];
  float t1 = 2.0f * T[0];
  int done = (t >= t1 - 1e-9f) ? 1 : 0;
  float dtc = done ? 1e-3f : fminf(dt, t1 - t);
  ctrl[2] = dtc; ctrl[3] = (float)done; ctrl[5] = t1;
  *errAcc = 0.0;
}

__global__ void combine_kernel(const float* __restrict__ y,
                               const float* __restrict__ k1, const float* __restrict__ k3,
                               const float* __restrict__ k4, const float* __restrict__ k5,
                               const float* __restrict__ k6,
                               const float* __restrict__ ctrl, float* __restrict__ y1) {
  int i = blockIdx.x * blockDim.x + threadIdx.x;
  if (i >= NY) return;
  const float B1 = (float)(35.0/384.0), B3 = (float)(500.0/1113.0);
  const float B4 = (float)(125.0/192.0), B5 = (float)(-2187.0/6784.0);
  const float B6 = (float)(11.0/84.0);
  float s = B1*k1[i] + B3*k3[i] + B4*k4[i] + B5*k5[i] + B6*k6[i];
  y1[i] = y[i] + ctrl[2] * s;
}

__global__ __launch_bounds__(256) void err_kernel(
    const float* __restrict__ y, const float* __restrict__ y1,
    const float* __restrict__ k1, const float* __restrict__ k3,
    const float* __restrict__ k4, const float* __restrict__ k5,
    const float* __restrict__ k6, const float* __restrict__ k7,
    const float* __restrict__ ctrl, double* __restrict__ errAcc) {
  __shared__ float red[256];
  int i = blockIdx.x * 256 + threadIdx.x;
  float v = 0.f;
  if (i < NY) {
    const float E1 = (float)(35.0/384.0 - 5179.0/57600.0);
    const float E3 = (float)(500.0/1113.0 - 7571.0/16695.0);
    const float E4 = (float)(125.0/192.0 - 393.0/640.0);
    const float E5 = (float)(-2187.0/6784.0 + 92097.0/339200.0);
    const float E6 = (float)(11.0/84.0 - 187.0/2100.0);
    const float E7 = (float)(-1.0/40.0);
    float ev = ctrl[2] * (E1*k1[i] + E3*k3[i] + E4*k4[i] + E5*k5[i] + E6*k6[i] + E7*k7[i]);
    float tol = 1e-5f + 1e-5f * fmaxf(fabsf(y[i]), fabsf(y1[i]));
    float r = ev / tol;
    v = r * r;
  }
  red[threadIdx.x] = v;
  __syncthreads();
  for (int s = 128; s > 0; s >>= 1) {
    if (threadIdx.x < s) red[threadIdx.x] += red[threadIdx.x + s];
    __syncthreads();
  }
  if (threadIdx.x == 0) atomicAdd(errAcc, (double)red[0]);
}

__global__ void step_finish_kernel(float* __restrict__ ctrl, const double* __restrict__ errAcc) {
  float t = ctrl[0], dt = ctrl[1], dtc = ctrl[2], t1 = ctrl[5];
  int done = ctrl[3] != 0.f;
  float errn = sqrtf((float)(*errAcc / (double)NY));
  int accept = (!done) && (errn <= 1.0f);
  if (accept) t += dtc;
  float factor = 0.9f * powf(errn, -0.2f);
  factor = fminf(fmaxf(factor, 0.2f), 10.f);
  float dtn = done ? dt : fminf(fmaxf(dtc * factor, 1e-6f), t1);
  ctrl[0] = t; ctrl[1] = dtn; ctrl[4] = (float)accept;
}

__global__ void commit_kernel(const float* __restrict__ ctrl,
                              const float* __restrict__ y1, float* __restrict__ y) {
  int i = blockIdx.x * blockDim.x + threadIdx.x;
  if (i < NY && ctrl[4] != 0.f) y[i] = y1[i];
}

__global__ void extract_kernel(const float* __restrict__ y, float* __restrict__ out) {
  int i = blockIdx.x * blockDim.x + threadIdx.x;
  if (i < N_ * Dd) out[i] = y[(i / Dd) * NYC + (i % Dd)];
}

// ---------------- host orchestration ----------------
extern "C" void kernel_launch(void* const* d_in, const int* in_sizes, int n_in,
                              void* d_out, int out_size, void* d_ws, size_t ws_size,
                              hipStream_t stream) {
  (void)in_sizes; (void)n_in; (void)out_size; (void)ws_size;
  const float* z  = (const float*)d_in[0];
  const float* Gw = (const float*)d_in[1];
  const float* W1 = (const float*)d_in[2];
  const float* b1 = (const float*)d_in[3];
  const float* W2 = (const float*)d_in[4];
  const float* b2 = (const float*)d_in[5];
  const float* W3 = (const float*)d_in[6];
  const float* b3 = (const float*)d_in[7];
  const float* T  = (const float*)d_in[8];

  float* wsf = (float*)d_ws;
  float* pre = wsf;
  float* y   = pre + PRE_FLOATS;
  float* y1  = y + NY;
  float* k[7];
  for (int i = 0; i < 7; ++i) k[i] = y1 + NY + (size_t)i * NY;
  float* ctrl = k[6] + NY;                    // 8 floats of control state
  double* errAcc = (double*)(ctrl + 8);       // 8-byte aligned by construction

  dim3 g1((NY + 255) / 256), b256(256);
  dim3 gf(N_ / 64), bf(128);

  precompute_kernel<<<1, 256, 0, stream>>>(Gw, W1, W2, W3, b3, pre);
  init_kernel<<<g1, b256, 0, stream>>>(z, y, ctrl);

  for (int step = 0; step < 10; ++step) {
    step_init_kernel<<<1, 1, 0, stream>>>(ctrl, errAcc, T);
    // k1 = f(y)
    f_kernel<<<gf, bf, 0, stream>>>(y, y, y, y, y, y, 0.f,0.f,0.f,0.f,0.f, 0,
                                    ctrl, W1, b1, W2, b2, pre, k[0]);
    f_kernel<<<gf, bf, 0, stream>>>(y, k[0], y, y, y, y,
                                    (float)(1.0/5.0), 0.f,0.f,0.f,0.f, 1,
                                    ctrl, W1, b1, W2, b2, pre, k[1]);
    f_kernel<<<gf, bf, 0, stream>>>(y, k[0], k[1], y, y, y,
                                    (float)(3.0/40.0), (float)(9.0/40.0), 0.f,0.f,0.f, 2,
                                    ctrl, W1, b1, W2, b2, pre, k[2]);
    f_kernel<<<gf, bf, 0, stream>>>(y, k[0], k[1], k[2], y, y,
                                    (float)(44.0/45.0), (float)(-56.0/15.0),
                                    (float)(32.0/9.0), 0.f, 0.f, 3,
                                    ctrl, W1, b1, W2, b2, pre, k[3]);
    f_kernel<<<gf, bf, 0, stream>>>(y, k[0], k[1], k[2], k[3], y,
                                    (float)(19372.0/6561.0), (float)(-25360.0/2187.0),
                                    (float)(64448.0/6561.0), (float)(-212.0/729.0), 0.f, 4,
                                    ctrl, W1, b1, W2, b2, pre, k[4]);
    f_kernel<<<gf, bf, 0, stream>>>(y, k[0], k[1], k[2], k[3], k[4],
                                    (float)(9017.0/3168.0), (float)(-355.0/33.0),
                                    (float)(46732.0/5247.0), (float)(49.0/176.0),
                                    (float)(-5103.0/18656.0), 5,
                                    ctrl, W1, b1, W2, b2, pre, k[5]);
    combine_kernel<<<g1, b256, 0, stream>>>(y, k[0], k[2], k[3], k[4], k[5], ctrl, y1);
    // k7 = f(y1)
    f_kernel<<<gf, bf, 0, stream>>>(y1, y, y, y, y, y, 0.f,0.f,0.f,0.f,0.f, 0,
                                    ctrl, W1, b1, W2, b2, pre, k[6]);
    err_kernel<<<g1, b256, 0, stream>>>(y, y1, k[0], k[2], k[3], k[4], k[5], k[6],
                                        ctrl, errAcc);
    step_finish_kernel<<<1, 1, 0, stream>>>(ctrl, errAcc);
    commit_kernel<<<g1, b256, 0, stream>>>(ctrl, y1, y);
  }
  extract_kernel<<<(N_ * Dd + 255) / 256, 256, 0, stream>>>(y, (float*)d_out);
}